// SG_4887672783508
// MI455X (gfx1250) — compile-verified
//
#include <hip/hip_runtime.h>
#include <math.h>

typedef float v2f __attribute__((ext_vector_type(2)));
typedef float v8f __attribute__((ext_vector_type(8)));

#define B_    16
#define N_    2048
#define DIN   128
#define CIN   256
#define COUT  256
#define S_    512
#define K_    32
#define M_    (B_ * S_)          // 8192 centers
#define NTOT  262144.0f          // M_*K_ samples per channel for BN stats
#define EPS_  1e-5f
#define LDAP  260                // padded LDS row (256 + 4) -> conflict-free b64 loads

// ---------------------------------------------------------------------------
// 1) transpose x [B,128,N] -> feats [B,N,128]
// ---------------------------------------------------------------------------
__global__ __launch_bounds__(256) void sg_transpose_x(const float* __restrict__ x,
                                                      float* __restrict__ feats) {
    size_t tid = (size_t)blockIdx.x * 256 + threadIdx.x;   // = (b*128+d)*2048 + n
    int n  = (int)(tid & 2047);
    int bd = (int)(tid >> 11);
    int d  = bd & 127;
    int b  = bd >> 7;
    feats[((size_t)(b * N_ + n)) * DIN + d] = x[tid];
}

// ---------------------------------------------------------------------------
// 2) FPS: one wave per batch, dists in registers, coords in LDS, shfl argmax
// ---------------------------------------------------------------------------
__global__ __launch_bounds__(32) void sg_fps(const float* __restrict__ coords,
                                             int* __restrict__ fps_idx,
                                             float* __restrict__ new_xyz) {
    __shared__ float cs[N_ * 3];
    const int b = blockIdx.x, lane = threadIdx.x;
    for (int i = lane; i < N_ * 3; i += 32) cs[i] = coords[(size_t)b * N_ * 3 + i];
    __syncthreads();

    float d[64];
#pragma unroll
    for (int i = 0; i < 64; ++i) d[i] = 1e10f;

    int far = 0;
    for (int s = 0; s < S_; ++s) {
        if (lane == 0) fps_idx[b * S_ + s] = far;
        float cx = cs[far * 3 + 0], cy = cs[far * 3 + 1], cz = cs[far * 3 + 2];
        if (lane < 3) new_xyz[((size_t)(b * S_ + s)) * 3 + lane] = cs[far * 3 + lane];

        float bv = -1.0f; int bn = 0x7fffffff;
#pragma unroll
        for (int i = 0; i < 64; ++i) {
            int n = i * 32 + lane;
            float dx = cs[n * 3 + 0] - cx;
            float dy = cs[n * 3 + 1] - cy;
            float dz = cs[n * 3 + 2] - cz;
            float dd = dx * dx + dy * dy + dz * dz;
            dd = fminf(d[i], dd);
            d[i] = dd;
            if (dd > bv) { bv = dd; bn = n; }   // strict > keeps earliest index per lane
        }
#pragma unroll
        for (int off = 16; off; off >>= 1) {    // argmax, tie -> smaller index (jnp.argmax)
            float ov = __shfl_xor(bv, off, 32);
            int   on = __shfl_xor(bn, off, 32);
            if (ov > bv || (ov == bv && on < bn)) { bv = ov; bn = on; }
        }
        far = bn;
    }
}

// ---------------------------------------------------------------------------
// 3) gather new_feat [M,128] = feats[b, fps_idx[m], :]
// ---------------------------------------------------------------------------
__global__ __launch_bounds__(128) void sg_gather_nf(const float* __restrict__ feats,
                                                    const int* __restrict__ fps_idx,
                                                    float* __restrict__ new_feat) {
    const int m = blockIdx.x, t = threadIdx.x;
    const int b = m >> 9;
    const int idx = fps_idx[m];
    new_feat[(size_t)m * DIN + t] = feats[((size_t)(b * N_ + idx)) * DIN + t];
}

// ---------------------------------------------------------------------------
// 4) kNN: one block per center; iterative 32x argmin over LDS distance array
// ---------------------------------------------------------------------------
__global__ __launch_bounds__(256) void sg_knn(const float* __restrict__ coords,
                                              const float* __restrict__ new_xyz,
                                              int* __restrict__ knn_idx) {
    __shared__ float d2s[N_];
    __shared__ float rv[256];
    __shared__ int   ri[256];
    const int m = blockIdx.x, t = threadIdx.x;
    const int b = m >> 9;
    const float cx = new_xyz[(size_t)m * 3 + 0];
    const float cy = new_xyz[(size_t)m * 3 + 1];
    const float cz = new_xyz[(size_t)m * 3 + 2];
#pragma unroll
    for (int i = 0; i < 8; ++i) {
        int n = i * 256 + t;
        const float* cp = coords + ((size_t)(b * N_ + n)) * 3;
        float dx = cp[0] - cx, dy = cp[1] - cy, dz = cp[2] - cz;
        d2s[n] = dx * dx + dy * dy + dz * dz;
    }
    __syncthreads();
    for (int kk = 0; kk < K_; ++kk) {
        float bv = 3.4e38f; int bi = 0x7fffffff;
#pragma unroll
        for (int i = 0; i < 8; ++i) {
            int n = i * 256 + t;
            float v = d2s[n];
            if (v < bv) { bv = v; bi = n; }
        }
        rv[t] = bv; ri[t] = bi;
        __syncthreads();
        for (int off = 128; off; off >>= 1) {
            if (t < off) {
                if (rv[t + off] < rv[t] || (rv[t + off] == rv[t] && ri[t + off] < ri[t])) {
                    rv[t] = rv[t + off]; ri[t] = ri[t + off];
                }
            }
            __syncthreads();
        }
        if (t == 0) { knn_idx[(size_t)m * K_ + kk] = ri[0]; d2s[ri[0]] = 3.4e38f; }
        __syncthreads();
    }
}

// ---------------------------------------------------------------------------
// 5) zero BN stat accumulators (re-done every launch -> graph-replay safe)
// ---------------------------------------------------------------------------
__global__ void sg_zero_stats(float* __restrict__ stats) { stats[threadIdx.x] = 0.0f; }

// ---------------------------------------------------------------------------
// GEMM core: per-block 256x32 = W(256x256) * G(256x32) via V_WMMA_F32_16X16X4_F32
// G tile is in LDS as [k][c] rows padded to LDAP. Each wave: 32 o-rows x 32 k.
// ---------------------------------------------------------------------------
__device__ __forceinline__ void sg_gemm_core(const float* __restrict__ W,
                                             const float* agg,
                                             float* __restrict__ yout,
                                             float* __restrict__ ssum,
                                             float* __restrict__ ssq) {
    const int lane = threadIdx.x & 31;
    const int wv   = threadIdx.x >> 5;   // 0..7
    const int half = lane >> 4;          // K-pair select
    const int col  = lane & 15;          // M for A-frag, N for B/D-frag

    v8f acc00 = {}; v8f acc01 = {}; v8f acc10 = {}; v8f acc11 = {};
    const int ob0 = wv * 32;
    const int ob1 = ob0 + 16;

#pragma unroll 4
    for (int c0 = 0; c0 < CIN; c0 += 4) {
        // A-frag (16x4): lanes 0-15 hold K={c0,c0+1}, lanes 16-31 K={c0+2,c0+3}
        const float* wp0 = W + (size_t)(ob0 + col) * CIN + c0 + 2 * half;
        const float* wp1 = W + (size_t)(ob1 + col) * CIN + c0 + 2 * half;
        v2f a0; a0.x = wp0[0]; a0.y = wp0[1];
        v2f a1; a1.x = wp1[0]; a1.y = wp1[1];
        // B-frag (4x16): row K striped across lanes; LDS [k][c] -> b64 per lane
        const float* gp0 = agg + (0  + col) * LDAP + c0 + 2 * half;
        const float* gp1 = agg + (16 + col) * LDAP + c0 + 2 * half;
        v2f b0; b0.x = gp0[0]; b0.y = gp0[1];
        v2f b1; b1.x = gp1[0]; b1.y = gp1[1];

        acc00 = __builtin_amdgcn_wmma_f32_16x16x4_f32(false, a0, false, b0, (short)0, acc00, false, false);
        acc01 = __builtin_amdgcn_wmma_f32_16x16x4_f32(false, a0, false, b1, (short)0, acc01, false, false);
        acc10 = __builtin_amdgcn_wmma_f32_16x16x4_f32(false, a1, false, b0, (short)0, acc10, false, false);
        acc11 = __builtin_amdgcn_wmma_f32_16x16x4_f32(false, a1, false, b1, (short)0, acc11, false, false);
    }

    // D layout: vgpr v: lanes 0-15 -> (M=v, N=col); lanes 16-31 -> (M=v+8, N=col)
#pragma unroll
    for (int v = 0; v < 8; ++v) {
        {
            int o = ob0 + v + 8 * half;
            float x0 = acc00[v], x1 = acc01[v];
            yout[(size_t)o * K_ + col]      = x0;
            yout[(size_t)o * K_ + 16 + col] = x1;
            float s = x0 + x1, q = x0 * x0 + x1 * x1;
#pragma unroll
            for (int mk = 1; mk < 16; mk <<= 1) { s += __shfl_xor(s, mk, 32); q += __shfl_xor(q, mk, 32); }
            if (col == 0) { atomicAdd(&ssum[o], s); atomicAdd(&ssq[o], q); }
        }
        {
            int o = ob1 + v + 8 * half;
            float x0 = acc10[v], x1 = acc11[v];
            yout[(size_t)o * K_ + col]      = x0;
            yout[(size_t)o * K_ + 16 + col] = x1;
            float s = x0 + x1, q = x0 * x0 + x1 * x1;
#pragma unroll
            for (int mk = 1; mk < 16; mk <<= 1) { s += __shfl_xor(s, mk, 32); q += __shfl_xor(q, mk, 32); }
            if (col == 0) { atomicAdd(&ssum[o], s); atomicAdd(&ssq[o], q); }
        }
    }
}

// ---------------------------------------------------------------------------
// 6) GEMM1: fused gather/center/concat -> LDS tile -> WMMA -> y1 + stats1
// ---------------------------------------------------------------------------
__global__ __launch_bounds__(256) void sg_gemm1(const float* __restrict__ feats,
                                                const float* __restrict__ new_feat,
                                                const int* __restrict__ knn_idx,
                                                const float* __restrict__ w1,
                                                float* __restrict__ ybuf,
                                                float* __restrict__ stats) {
    __shared__ float agg[K_ * LDAP];
    const int m = blockIdx.x;
    const int b = m >> 9;
    const int t = threadIdx.x;
    const int k = t >> 3;
    const int seg = t & 7;
    const int c0 = seg * 32;
    const int nk = knn_idx[(size_t)m * K_ + k];
    const float* fp = feats + ((size_t)(b * N_ + nk)) * DIN;
    const float* nf = new_feat + (size_t)m * DIN;
    if (seg < 4) {
#pragma unroll
        for (int j = 0; j < 32; ++j) { int c = c0 + j; agg[k * LDAP + c] = fp[c] - nf[c]; }
    } else {
#pragma unroll
        for (int j = 0; j < 32; ++j) { int c = c0 + j; agg[k * LDAP + c] = nf[c - 128]; }
    }
    __syncthreads();
    sg_gemm_core(w1, agg, ybuf + (size_t)m * (COUT * K_), stats + 0, stats + 256);
}

// ---------------------------------------------------------------------------
// 7) GEMM2: BN1+ReLU on the fly -> LDS tile -> WMMA -> y2 (in place) + stats2
// ---------------------------------------------------------------------------
__global__ __launch_bounds__(256) void sg_gemm2(const float* __restrict__ w2,
                                                const float* __restrict__ gamma1,
                                                const float* __restrict__ beta1,
                                                float* __restrict__ ybuf,
                                                float* __restrict__ stats) {
    __shared__ float agg[K_ * LDAP];
    const int m = blockIdx.x;
    const int t = threadIdx.x;     // input channel c
    const float invNT = 1.0f / NTOT;
    float mean = stats[t] * invNT;
    float var  = stats[256 + t] * invNT - mean * mean;
    float sc = gamma1[t] * rsqrtf(var + EPS_);
    float sh = beta1[t] - mean * sc;
    float* yp = ybuf + (size_t)m * (COUT * K_) + (size_t)t * K_;
#pragma unroll
    for (int k = 0; k < K_; ++k) agg[k * LDAP + t] = fmaxf(yp[k] * sc + sh, 0.0f);
    __syncthreads();
    sg_gemm_core(w2, agg, ybuf + (size_t)m * (COUT * K_), stats + 512, stats + 768);
}

// ---------------------------------------------------------------------------
// 8) finalize: BN2 + ReLU + max over k -> out [B,256,S]
// ---------------------------------------------------------------------------
__global__ __launch_bounds__(256) void sg_finalize(const float* __restrict__ ybuf,
                                                   const float* __restrict__ stats,
                                                   const float* __restrict__ gamma2,
                                                   const float* __restrict__ beta2,
                                                   float* __restrict__ out) {
    const int m = blockIdx.x, o = threadIdx.x;
    const int b = m >> 9, s = m & 511;
    const float invNT = 1.0f / NTOT;
    float mean = stats[512 + o] * invNT;
    float var  = stats[768 + o] * invNT - mean * mean;
    float sc = gamma2[o] * rsqrtf(var + EPS_);
    float sh = beta2[o] - mean * sc;
    const float* yp = ybuf + (size_t)m * (COUT * K_) + (size_t)o * K_;
    float mx = 0.0f;   // ReLU outputs are >= 0, so 0 is a valid identity for max
#pragma unroll
    for (int k = 0; k < K_; ++k) mx = fmaxf(mx, fmaxf(yp[k] * sc + sh, 0.0f));
    out[(size_t)(B_ * S_ * 3) + ((size_t)(b * COUT + o)) * S_ + s] = mx;
}

// ---------------------------------------------------------------------------
extern "C" void kernel_launch(void* const* d_in, const int* in_sizes, int n_in,
                              void* d_out, int out_size, void* d_ws, size_t ws_size,
                              hipStream_t stream) {
    const float* x      = (const float*)d_in[0];
    const float* coords = (const float*)d_in[1];
    const float* w1     = (const float*)d_in[2];
    const float* gamma1 = (const float*)d_in[3];
    const float* beta1  = (const float*)d_in[4];
    const float* w2     = (const float*)d_in[5];
    const float* gamma2 = (const float*)d_in[6];
    const float* beta2  = (const float*)d_in[7];
    float* out = (float*)d_out;

    // workspace layout
    float* feats    = (float*)d_ws;                                  // 4,194,304 f
    float* new_feat = feats + (size_t)B_ * N_ * DIN;                 // 1,048,576 f
    int*   fps_idx  = (int*)(new_feat + (size_t)M_ * DIN);           // 8,192 i
    int*   knn_idx  = fps_idx + M_;                                  // 262,144 i
    float* stats    = (float*)(knn_idx + (size_t)M_ * K_);           // 1,024 f
    float* ybuf     = stats + 1024;                                  // 67,108,864 f

    float* new_xyz = out;   // first B*S*3 floats of output

    sg_transpose_x<<<(B_ * N_ * DIN) / 256, 256, 0, stream>>>(x, feats);
    sg_fps<<<B_, 32, 0, stream>>>(coords, fps_idx, new_xyz);
    sg_gather_nf<<<M_, 128, 0, stream>>>(feats, fps_idx, new_feat);
    sg_knn<<<M_, 256, 0, stream>>>(coords, new_xyz, knn_idx);
    sg_zero_stats<<<1, 1024, 0, stream>>>(stats);
    sg_gemm1<<<M_, 256, 0, stream>>>(feats, new_feat, knn_idx, w1, ybuf, stats);
    sg_gemm2<<<M_, 256, 0, stream>>>(w2, gamma1, beta1, ybuf, stats);
    sg_finalize<<<M_, 256, 0, stream>>>(ybuf, stats, gamma2, beta2, out);
}